// PillarFocusScatter_23381801959694
// MI455X (gfx1250) — compile-verified
//
#include <hip/hip_runtime.h>
#include <math.h>

typedef __attribute__((ext_vector_type(2))) float v2f;
typedef __attribute__((ext_vector_type(8))) float v8f;

#define NXg 432
#define NYg 496
#define Cg 64
#define PLANE (NXg * NYg)   /* 214272 */
#define GRIDg PLANE
#define ROWS 68             /* LDS row stride (floats), bank-conflict pad */

__global__ __launch_bounds__(256) void pillar_zero_kernel(float* __restrict__ out, long long n) {
    long long n4 = n >> 2;
    float4 z = make_float4(0.f, 0.f, 0.f, 0.f);
    long long i = (long long)blockIdx.x * blockDim.x + threadIdx.x;
    long long stride = (long long)gridDim.x * blockDim.x;
    float4* out4 = (float4*)out;
    for (; i < n4; i += stride) out4[i] = z;
    // tail (out_size is a multiple of 64 in practice, but be safe)
    if (blockIdx.x == 0 && threadIdx.x == 0) {
        for (long long t = n4 << 2; t < n; ++t) out[t] = 0.f;
    }
}

__global__ __launch_bounds__(256) void pillar_scatter_attn_kernel(
    const float* __restrict__ pf,        // (P, 64) f32
    const long long* __restrict__ coords,// (P, 4) i64: b, z, y, x
    const float* __restrict__ w1,        // (16, 64)
    const float* __restrict__ b1,        // (16,)
    const float* __restrict__ w2,        // (1, 16)
    const float* __restrict__ b2,        // (1,)
    float* __restrict__ out,             // (B, 64, NY, NX)
    int P)
{
    __shared__ float w1s[16 * ROWS];
    __shared__ float tile[8][16 * ROWS];

    const int tid  = threadIdx.x;
    const int lane = tid & 31;
    const int wv   = tid >> 5;

    // Stage w1 (16x64) into LDS, padded row stride
    for (int i = tid; i < 16 * 64; i += 256) {
        int o = i >> 6, k = i & 63;
        w1s[o * ROWS + k] = w1[i];
    }

    const int waveBase = (blockIdx.x * 8 + wv) * 16;

    // Stage 16 pillar feature rows (coalesced) into this wave's LDS tile
    for (int j = 0; j < 16; ++j) {
        int p = waveBase + j;
        float f0 = 0.f, f1 = 0.f;
        if (p < P) {
            const float* row = pf + (long long)p * Cg;
            f0 = row[lane];
            f1 = row[lane + 32];
        }
        tile[wv][j * ROWS + lane]      = f0;
        tile[wv][j * ROWS + lane + 32] = f1;
    }

    // Lanes 0..15 each compute the output base offset for pillar (waveBase+lane)
    int myBase = -1;
    if (lane < 16) {
        int p = waveBase + lane;
        if (p < P) {
            const long long* c = coords + (long long)p * 4;
            long long b = c[0];
            long long y = c[2];
            long long x = c[3];
            long long idx = y * NXg + x;          // z == 0
            if (idx >= 0 && idx < GRIDg)
                myBase = (int)(b * (long long)Cg * PLANE + idx);
        }
    }
    __syncthreads();

    // H(16x16) = F(16x64) * W1^T(64x16) via 16 chained f32 WMMA 16x16x4 steps.
    const int row  = lane & 15;   // A: M row / B: N column handled by this lane
    const int half = lane >> 4;   // K sub-pair selector
    v8f acc = {0.f, 0.f, 0.f, 0.f, 0.f, 0.f, 0.f, 0.f};
    #pragma unroll
    for (int kk = 0; kk < 16; ++kk) {
        int col = kk * 4 + half * 2;
        v2f a = *(const v2f*)&tile[wv][row * ROWS + col]; // F[row][col], F[row][col+1]
        v2f b = *(const v2f*)&w1s[row * ROWS + col];      // W1[n][col],  W1[n][col+1]
        acc = __builtin_amdgcn_wmma_f32_16x16x4_f32(
            /*neg_a=*/false, a, /*neg_b=*/false, b,
            /*c_mod=*/(short)0, acc, /*reuse_a=*/false, /*reuse_b=*/false);
    }

    // Epilogue: relu(+b1), dot with w2 (cross-lane reduce over the 16 N-lanes), sigmoid.
    const float b1o = b1[row];
    const float w2o = w2[row];
    const float b2s = b2[0];
    float att[8];
    #pragma unroll
    for (int j = 0; j < 8; ++j) {
        float h = acc[j] + b1o;            // C layout: VGPR j -> M = j + 8*half, N = row
        h = h > 0.f ? h : 0.f;
        float part = h * w2o;
        part += __shfl_xor(part, 1, 32);
        part += __shfl_xor(part, 2, 32);
        part += __shfl_xor(part, 4, 32);
        part += __shfl_xor(part, 8, 32);   // stays within 16-lane half
        att[j] = 1.f / (1.f + __expf(-(part + b2s)));
    }

    // Scatter: out[b, c, y, x] = f[c] * att  (channel-major planes, stride PLANE)
    #pragma unroll
    for (int p = 0; p < 16; ++p) {
        int   srcHalf = (p < 8) ? 0 : 16;               // att for pillar p lives in slot p&7
        float attp  = __shfl(att[p & 7], srcHalf, 32);  // (lanes 0-15: M=slot, 16-31: M=slot+8)
        int   basep = __shfl(myBase, p, 32);            // base computed by lane p
        if (basep >= 0) {
            float f0 = tile[wv][p * ROWS + lane];
            float f1 = tile[wv][p * ROWS + lane + 32];
            out[(long long)basep + (long long)lane * PLANE]        = f0 * attp;
            out[(long long)basep + (long long)(lane + 32) * PLANE] = f1 * attp;
        }
    }
}

extern "C" void kernel_launch(void* const* d_in, const int* in_sizes, int n_in,
                              void* d_out, int out_size, void* d_ws, size_t ws_size,
                              hipStream_t stream) {
    const float*     pf     = (const float*)d_in[0];
    const long long* coords = (const long long*)d_in[1];
    const float*     w1     = (const float*)d_in[2];
    const float*     b1     = (const float*)d_in[3];
    const float*     w2     = (const float*)d_in[4];
    const float*     b2     = (const float*)d_in[5];
    float* out = (float*)d_out;

    const int P = in_sizes[0] / Cg;

    // 1) Zero the dense output (only scattered pillar cells are non-zero).
    pillar_zero_kernel<<<8192, 256, 0, stream>>>(out, (long long)out_size);

    // 2) Fused per-pillar MLP (WMMA) + attention + scatter. 128 pillars/block.
    int blocks = (P + 127) / 128;
    if (blocks < 1) blocks = 1;
    pillar_scatter_attn_kernel<<<blocks, 256, 0, stream>>>(pf, coords, w1, b1, w2, b2, out, P);
}